// StationaryWaveletTransform_64372969832979
// MI455X (gfx1250) — compile-verified
//
#include <hip/hip_runtime.h>

// Stationary (a trous) Haar wavelet transform, 3 levels, circular boundary,
// fused across levels:
//   cA1[i] = s*(x[i]+x[i+1])                cD1[i] = s*(x[i]-x[i+1])
//   cA2[i] = s2*(x[i]+x[i+1]+x[i+2]+x[i+3]) cD2[i] = s2*(x[i]+x[i+1]-x[i+2]-x[i+3])
//   cA3[i] = s3*sum(x[i..i+7])              cD3[i] = s3*(sum(x[i..i+3])-sum(x[i+4..i+7]))
// with s = 1/sqrt(2), s2 = s*s, s3 = s*s*s, indices mod N.
// Output layout (coarsest first, pairs stacked): [cA3,cD3, cA2,cD2, cA1,cD1],
// each plane is rows*N floats.

typedef float v4f __attribute__((ext_vector_type(4)));
typedef int   v4i __attribute__((vector_size(16)));   // matches builtin param type

#define SWT_N       8192
#define SWT_THREADS 256
#define SWT_HALO    8
#define SWT_GROUPS  (SWT_N / (4 * SWT_THREADS))  // 8 float4-groups per thread

#if __has_builtin(__builtin_amdgcn_global_load_async_to_lds_b128) && \
    __has_builtin(__builtin_amdgcn_s_wait_asynccnt)
#define SWT_ASYNC 1
#else
#define SWT_ASYNC 0
#endif

__global__ __launch_bounds__(SWT_THREADS) void
swt_haar3_kernel(const float* __restrict__ x, float* __restrict__ out, int rows) {
  __shared__ float s_x[SWT_N + SWT_HALO];

  const int tid = (int)threadIdx.x;
  const int row = (int)blockIdx.x;
  const float* __restrict__ xr = x + (size_t)row * SWT_N;

  // ---- Stage the row (plus 8-element circular halo) into LDS ----
#if SWT_ASYNC
  {
    __attribute__((address_space(1))) char* gsrc =
        (__attribute__((address_space(1))) char*)(const_cast<float*>(xr));
    __attribute__((address_space(3))) char* ldst =
        (__attribute__((address_space(3))) char*)s_x;
#pragma unroll
    for (int k = 0; k < SWT_GROUPS; ++k) {
      const int byteOff = (tid + k * SWT_THREADS) * 16;
      __builtin_amdgcn_global_load_async_to_lds_b128(
          (__attribute__((address_space(1))) v4i*)(gsrc + byteOff),
          (__attribute__((address_space(3))) v4i*)(ldst + byteOff), 0, 0);
    }
    if (tid < SWT_HALO / 4) {
      __builtin_amdgcn_global_load_async_to_lds_b128(
          (__attribute__((address_space(1))) v4i*)(gsrc + tid * 16),
          (__attribute__((address_space(3))) v4i*)(ldst + SWT_N * 4 + tid * 16),
          0, 0);
    }
    __builtin_amdgcn_s_wait_asynccnt(0);
  }
#else
  {
#pragma unroll
    for (int k = 0; k < SWT_GROUPS; ++k) {
      const int idx = tid + k * SWT_THREADS;
      v4f v = __builtin_nontemporal_load((const v4f*)xr + idx);
      *(v4f*)(s_x + idx * 4) = v;
    }
    if (tid < SWT_HALO / 4) {
      v4f v = __builtin_nontemporal_load((const v4f*)xr + tid);
      *(v4f*)(s_x + SWT_N + tid * 4) = v;
    }
  }
#endif
  __syncthreads();

  const float S1 = 0.70710678118654752440f;  // 1/sqrt(2)
  const float S2 = 0.5f;                     // (1/sqrt(2))^2
  const float S3 = S1 * S2;                  // (1/sqrt(2))^3

  const size_t plane = (size_t)rows * SWT_N;
  const size_t rb    = (size_t)row * SWT_N;
  float* __restrict__ oA3 = out + 0 * plane + rb;
  float* __restrict__ oD3 = out + 1 * plane + rb;
  float* __restrict__ oA2 = out + 2 * plane + rb;
  float* __restrict__ oD2 = out + 3 * plane + rb;
  float* __restrict__ oA1 = out + 4 * plane + rb;
  float* __restrict__ oD1 = out + 5 * plane + rb;

#pragma unroll
  for (int k = 0; k < SWT_GROUPS; ++k) {
    const int b = tid * 4 + k * (4 * SWT_THREADS);  // 16B-aligned, conflict-free

    v4f p0 = *(const v4f*)(s_x + b);
    v4f p1 = *(const v4f*)(s_x + b + 4);
    v4f p2 = *(const v4f*)(s_x + b + 8);
    float v[12] = {p0.x, p0.y, p0.z, p0.w,
                   p1.x, p1.y, p1.z, p1.w,
                   p2.x, p2.y, p2.z, p2.w};

    v4f a1, d1, a2, d2, a3, d3;
#pragma unroll
    for (int e = 0; e < 4; ++e) {
      const float s01 = v[e + 0] + v[e + 1];
      const float m01 = v[e + 0] - v[e + 1];
      const float s23 = v[e + 2] + v[e + 3];
      const float s45 = v[e + 4] + v[e + 5];
      const float s67 = v[e + 6] + v[e + 7];
      const float q0  = s01 + s23;
      const float q1  = s45 + s67;
      a1[e] = s01 * S1;
      d1[e] = m01 * S1;
      a2[e] = q0 * S2;
      d2[e] = (s01 - s23) * S2;
      a3[e] = (q0 + q1) * S3;
      d3[e] = (q0 - q1) * S3;
    }

    // Write-once, streaming output larger than L2: non-temporal b128 stores.
    __builtin_nontemporal_store(a3, (v4f*)(oA3 + b));
    __builtin_nontemporal_store(d3, (v4f*)(oD3 + b));
    __builtin_nontemporal_store(a2, (v4f*)(oA2 + b));
    __builtin_nontemporal_store(d2, (v4f*)(oD2 + b));
    __builtin_nontemporal_store(a1, (v4f*)(oA1 + b));
    __builtin_nontemporal_store(d1, (v4f*)(oD1 + b));
  }
}

extern "C" void kernel_launch(void* const* d_in, const int* in_sizes, int n_in,
                              void* d_out, int out_size, void* d_ws, size_t ws_size,
                              hipStream_t stream) {
  (void)n_in; (void)out_size; (void)d_ws; (void)ws_size;
  const float* x = (const float*)d_in[0];
  float* out = (float*)d_out;
  const int rows = in_sizes[0] / SWT_N;  // 32*64 = 2048
  if (rows <= 0) return;
  swt_haar3_kernel<<<rows, SWT_THREADS, 0, stream>>>(x, out, rows);
}